// KNNSpaceMean_53472342835586
// MI455X (gfx1250) — compile-verified
//
#include <hip/hip_runtime.h>

typedef __attribute__((ext_vector_type(2))) float v2f;
typedef __attribute__((ext_vector_type(8))) float v8f;

#define NBATCH 4
#define NPTS   8000
#define NCH    256
#define CHUNK  2048              // candidate points per staged chunk
#define NCHUNK 4                 // ceil(8000/2048)
#define WAVES  16
#define BLOCK  (WAVES * 32)      // 512 threads = 16 wave32
#define QTILE  16                // queries per wave (one WMMA N-dim tile)
#define QPB    (WAVES * QTILE)   // 256 queries per block
#define TILES_PER_BATCH 32       // ceil(8000/256)

__device__ __forceinline__ void wait_asynccnt0() {
#if defined(__has_builtin) && __has_builtin(__builtin_amdgcn_s_wait_asynccnt)
    __builtin_amdgcn_s_wait_asynccnt(0);
#else
    asm volatile("s_wait_asynccnt 0x0" ::: "memory");
#endif
}

// Async DMA one float (b32) from global (saddr64 + voff bytes) into LDS at ldsoff.
__device__ __forceinline__ void async_g2lds_b32(unsigned ldsoff, unsigned voff,
                                                unsigned long long saddr) {
    asm volatile("global_load_async_to_lds_b32 %0, %1, %2"
                 :: "v"(ldsoff), "v"(voff), "s"(saddr) : "memory");
}

__global__ __launch_bounds__(BLOCK)
void knn2_mean_kernel(const float* __restrict__ points,
                      const float* __restrict__ preds,
                      float* __restrict__ out)
{
    // Double-buffered raw candidate coords (x,y,z stride-3): 2 x 24 KB
    __shared__ float s_raw[2][CHUNK * 3];

    const int b     = blockIdx.x >> 5;             // /32 blocks per batch
    const int qblk  = (blockIdx.x & 31) * QPB;
    const int wave  = threadIdx.x >> 5;
    const int lane  = threadIdx.x & 31;
    const int half  = lane >> 4;                   // 0: lanes 0-15, 1: lanes 16-31
    const int l16   = lane & 15;
    const int qbase = qblk + wave * QTILE;

    const float* __restrict__ pb = points + (size_t)b * NPTS * 3;
    const unsigned long long pb64 = (unsigned long long)(uintptr_t)pb;

    // ---- B matrix operand: query (x,y,z,1).
    // B[k][n] layout: lanes 0-15 hold (K0,K1)=(xq,yq); lanes 16-31 hold (K2,K3)=(zq,1)
    int q  = qbase + l16;
    int qc = (q < NPTS) ? q : (NPTS - 1);          // clamp: keeps EXEC all-ones for WMMA
    float qx = pb[qc * 3 + 0];
    float qy = pb[qc * 3 + 1];
    float qz = pb[qc * 3 + 2];
    v2f bmat;
    bmat.x = half ? qz : qx;
    bmat.y = half ? 1.0f : qy;

    // Per-lane running top-2 of score = dot(q,c) - 0.5*|c|^2  (argmax == nearest)
    float best  = -3.0e38f, best2 = -3.0e38f;
    int   bi    = 0,        bi2   = 0;

    // ---- Prologue: async-stage chunk 0 into buffer 0
    {
        const int n0 = ((NPTS < CHUNK) ? NPTS : CHUNK) * 3;
        for (int i = threadIdx.x; i < n0; i += BLOCK) {
            unsigned ldsoff = (unsigned)(uintptr_t)&s_raw[0][i];
            async_g2lds_b32(ldsoff, (unsigned)(i * 4), pb64);
        }
        wait_asynccnt0();
        __syncthreads();
    }

    for (int k = 0; k < NCHUNK; ++k) {
        const int base = k * CHUNK;
        const int sel  = k & 1;

        // ---- Kick off async staging of chunk k+1 while we crunch chunk k
        if (k + 1 < NCHUNK) {
            const int nbase = (k + 1) * CHUNK;
            const int nn = (((NPTS - nbase) < CHUNK) ? (NPTS - nbase) : CHUNK) * 3;
            const int nsel = sel ^ 1;
            for (int i = threadIdx.x; i < nn; i += BLOCK) {
                unsigned ldsoff = (unsigned)(uintptr_t)&s_raw[nsel][i];
                async_g2lds_b32(ldsoff, (unsigned)((nbase * 3 + i) * 4), pb64);
            }
        }

        // ---- Scan chunk: one V_WMMA_F32_16X16X4_F32 per 16 candidates.
        // NPTS % 16 == 0, so tiles are always full: no masking needed.
        const int lim = ((NPTS - base) < CHUNK) ? (NPTS - base) : CHUNK;
        for (int t = 0; t < lim; t += 16) {
            const int c = t + l16;
            const float* rp = &s_raw[sel][c * 3];
            float x = rp[0], y = rp[1], z = rp[2];
            float s = -0.5f * (x * x + y * y + z * z);

            // A layout: lanes 0-15 = (x,y) of cand; lanes 16-31 = (z, s)
            v2f amat;
            amat.x = half ? z : x;
            amat.y = half ? s : y;

            v8f d = {};
            d = __builtin_amdgcn_wmma_f32_16x16x4_f32(
                    /*neg_a=*/false, amat, /*neg_b=*/false, bmat,
                    /*c_mod=*/(short)0, d, /*reuse_a=*/false, /*reuse_b=*/false);

            // D layout: lane owns query l16; VGPR r = candidate t + r + 8*half
            const int candBase = base + t + (half << 3);
#pragma unroll
            for (int r = 0; r < 8; ++r) {
                float v  = d[r];
                int   ci = candBase + r;
                bool  a1 = v > best;
                bool  a2 = v > best2;
                float t2 = a1 ? best : v;
                int   ti = a1 ? bi   : ci;
                best2 = a2 ? t2 : best2;
                bi2   = a2 ? ti : bi2;
                best  = a1 ? v  : best;
                bi    = a1 ? ci : bi;
            }
        }

        // Own async DMAs done BEFORE barrier -> after barrier all waves' data landed
        wait_asynccnt0();
        __syncthreads();
    }

    // ---- Merge the two half-wave candidate partitions (lane L <-> L+16)
    float ob  = __shfl_xor(best,  16, 32);
    float ob2 = __shfl_xor(best2, 16, 32);
    int   oi  = __shfl_xor(bi,    16, 32);
    int   oi2 = __shfl_xor(bi2,   16, 32);
    int i1, i2;
    if (ob > best) {
        i1 = oi;
        i2 = (best > ob2) ? bi : oi2;
    } else {
        i1 = bi;
        i2 = (ob > best2) ? oi : bi2;
    }

    // ---- Output: out[b][q][:] = 0.5*(preds[b][i1][:] + preds[b][i2][:])
    const float* __restrict__ predb = preds + (size_t)b * NPTS * NCH;
    float* __restrict__       outb  = out   + (size_t)b * NPTS * NCH;

    for (int j = 0; j < QTILE; ++j) {
        int qj = qbase + j;
        if (qj >= NPTS) break;                    // uniform across the wave
        int n0 = __shfl(i1, j, 32);
        int n1 = __shfl(i2, j, 32);
        const float4* p0 = (const float4*)(predb + (size_t)n0 * NCH);
        const float4* p1 = (const float4*)(predb + (size_t)n1 * NCH);
        float4*       o  = (float4*)(outb + (size_t)qj * NCH);
#pragma unroll
        for (int tt = 0; tt < 2; ++tt) {
            int c = lane + tt * 32;               // 64 float4 per 256-ch row
            float4 a = p0[c];
            float4 bb = p1[c];
            float4 r;
            r.x = 0.5f * (a.x + bb.x);
            r.y = 0.5f * (a.y + bb.y);
            r.z = 0.5f * (a.z + bb.z);
            r.w = 0.5f * (a.w + bb.w);
            o[c] = r;
        }
    }
}

extern "C" void kernel_launch(void* const* d_in, const int* in_sizes, int n_in,
                              void* d_out, int out_size, void* d_ws, size_t ws_size,
                              hipStream_t stream) {
    (void)in_sizes; (void)n_in; (void)out_size; (void)d_ws; (void)ws_size;
    const float* points = (const float*)d_in[0];   // [4, 8000, 3]
    const float* preds  = (const float*)d_in[1];   // [4, 8000, 256]
    // d_in[2] (k_vector) is dead in the reference: k is hardcoded to 2.
    float* out = (float*)d_out;                    // [4, 8000, 256]

    dim3 grid(NBATCH * TILES_PER_BATCH);           // 128 blocks
    dim3 block(BLOCK);                             // 512 threads = 16 wave32
    knn2_mean_kernel<<<grid, block, 0, stream>>>(points, preds, out);
}